// CoTr_61108794687987
// MI455X (gfx1250) — compile-verified
//
#include <hip/hip_runtime.h>

typedef __attribute__((ext_vector_type(2))) float v2f;
typedef __attribute__((ext_vector_type(8))) float v8f;

#define Hs 48
#define Ws 48
#define Ts 48
#define Cs 32
#define TILE_T 14   // usable voxel rows per 16-row WMMA tile (band needs cols m..m+2)

__global__ __launch_bounds__(32)
void natt_disp_kernel(const float* __restrict__ q,
                      const float* __restrict__ k,
                      float* __restrict__ out) {
    __shared__ float sD[16 * 16];   // one 16x16 f32 score tile, single wave per block

    const int lane = threadIdx.x;   // 0..31, wave32
    const int m    = lane & 15;     // row (A) / column (B,D) index
    const int hi   = lane >> 4;     // half-wave selector (K-pair 0..1 vs 2..3)
    const int h    = blockIdx.z;
    const int w    = blockIdx.y;
    const int t0   = blockIdx.x * TILE_T;

    // ---- A fragments: Q rows are voxels t0+m, K dim is channels ----
    int ta = t0 + m; if (ta > Ts - 1) ta = Ts - 1;   // clamp: rows >= TILE_T are don't-care
    const float* qrow = q + (size_t)(((h * Ws + w) * Ts + ta) * Cs);
    v2f a[8];
#pragma unroll
    for (int kc = 0; kc < 8; ++kc)
        a[kc] = *(const v2f*)(qrow + 4 * kc + 2 * hi);

    float s[27];

    // B column m holds key at t = t0-1+m (zero outside the volume -> score 0, like jnp.pad)
    const int  kt   = t0 - 1 + m;
    const bool t_ok = (unsigned)kt < (unsigned)Ts;

    for (int j9 = 0; j9 < 9; ++j9) {
        const int dh = j9 / 3 - 1;
        const int dw = j9 % 3 - 1;
        const int hh = h + dh;
        const int ww = w + dw;
        const bool ok = t_ok && (unsigned)hh < (unsigned)Hs && (unsigned)ww < (unsigned)Ws;

        v2f b[8];
        v2f zero = {};
        if (ok) {
            const float* krow = k + (size_t)(((hh * Ws + ww) * Ts + kt) * Cs);
#pragma unroll
            for (int kc = 0; kc < 8; ++kc)
                b[kc] = *(const v2f*)(krow + 4 * kc + 2 * hi);
        } else {
#pragma unroll
            for (int kc = 0; kc < 8; ++kc)
                b[kc] = zero;
        }

        // D[m_row, n_col] = sum_c Q[t0+m_row, c] * K[hh, ww, t0-1+n_col, c]
        v8f acc = {};
#pragma unroll
        for (int kc = 0; kc < 8; ++kc)
            acc = __builtin_amdgcn_wmma_f32_16x16x4_f32(
                /*neg_a=*/false, a[kc], /*neg_b=*/false, b[kc],
                /*c_mod=*/(short)0, acc, /*reuse_a=*/false, /*reuse_b=*/false);

        __syncthreads();            // prior iteration's band reads done before overwrite
#pragma unroll
        for (int r = 0; r < 8; ++r)                     // VGPR r: row r (+8 for hi half)
            sD[(r + 8 * hi) * 16 + m] = acc[r];
        __syncthreads();

        if (lane < TILE_T) {        // lane owns voxel t0+lane; band = cols lane..lane+2
#pragma unroll
            for (int dt = 0; dt < 3; ++dt)
                s[j9 * 3 + dt] = sD[lane * 16 + lane + dt];   // j = dh_i*9 + dw_i*3 + dt_i
        }
    }

    // ---- softmax over 27 + displacement-weighted sum ----
    const int t = t0 + lane;
    if (lane < TILE_T && t < Ts) {
        float mx = s[0];
#pragma unroll
        for (int j = 1; j < 27; ++j) mx = fmaxf(mx, s[j]);
        float sum = 0.f, oh = 0.f, ow = 0.f, ot = 0.f;
#pragma unroll
        for (int j = 0; j < 27; ++j) {
            float e = __expf(s[j] - mx);
            sum += e;
            oh += e * (float)(j / 9 - 1);
            ow += e * (float)((j / 3) % 3 - 1);
            ot += e * (float)(j % 3 - 1);
        }
        const float inv   = 1.0f / sum;
        const int   base  = (h * Ws + w) * Ts + t;
        const int   plane = Hs * Ws * Ts;
        out[0 * plane + base] = oh * inv;   // layout [B,3,H,W,T]
        out[1 * plane + base] = ow * inv;
        out[2 * plane + base] = ot * inv;
    }
}

extern "C" void kernel_launch(void* const* d_in, const int* in_sizes, int n_in,
                              void* d_out, int out_size, void* d_ws, size_t ws_size,
                              hipStream_t stream) {
    const float* q = (const float*)d_in[0];
    const float* k = (const float*)d_in[1];
    float* out     = (float*)d_out;
    (void)in_sizes; (void)n_in; (void)out_size; (void)d_ws; (void)ws_size;

    dim3 grid((Ts + TILE_T - 1) / TILE_T, Ws, Hs);   // (4, 48, 48) -> 9216 waves
    dim3 block(32);                                   // one wave32 per block
    natt_disp_kernel<<<grid, block, 0, stream>>>(q, k, out);
}